// NameEmbed_3908420239567
// MI455X (gfx1250) — compile-verified
//
#include <hip/hip_runtime.h>

#define N_NODES 100000
#define N_EDGES 1600000
#define DIM     128
#define N_SEEDS 10000

typedef __attribute__((ext_vector_type(2))) float v2f;
typedef __attribute__((ext_vector_type(8))) float v8f;

static inline int cdiv(long long a, int b) { return (int)((a + b - 1) / b); }

// deg[i] = 1 (self loop)
__global__ void k_deg_init(float* __restrict__ deg) {
  int i = blockIdx.x * blockDim.x + threadIdx.x;
  if (i < N_NODES) deg[i] = 1.0f;
}

// symmetrized degree: each directed edge (u,v) contributes to deg[v] and deg[u]
__global__ void k_deg_count(const int* __restrict__ e0, const int* __restrict__ e1,
                            float* __restrict__ deg) {
  int e = blockIdx.x * blockDim.x + threadIdx.x;
  if (e < N_EDGES) {
    atomicAdd(&deg[e0[e]], 1.0f);
    atomicAdd(&deg[e1[e]], 1.0f);
  }
}

// in-place deg -> rsqrt(deg)
__global__ void k_rsqrt(float* __restrict__ deg) {
  int i = blockIdx.x * blockDim.x + threadIdx.x;
  if (i < N_NODES) deg[i] = rsqrtf(deg[i]);
}

// self-loop contribution: agg[i] = h[i] * inv[i]^2   (also initializes agg)
__global__ void k_agg_init(float* __restrict__ agg, const float* __restrict__ h,
                           const float* __restrict__ inv) {
  int idx = blockIdx.x * blockDim.x + threadIdx.x;   // N * (DIM/4) chunks
  if (idx < N_NODES * (DIM / 4)) {
    int i = idx >> 5;
    int c = (idx & 31) * 4;
    float s = inv[i]; s = s * s;
    float4 hv = *(const float4*)(h + (size_t)i * DIM + c);
    float4 o  = { hv.x * s, hv.y * s, hv.z * s, hv.w * s };
    *(float4*)(agg + (size_t)i * DIM + c) = o;
  }
}

// one wave per edge; each lane owns a float4 chunk; both directions (u->v, v->u)
__global__ void k_scatter(float* __restrict__ agg, const float* __restrict__ h,
                          const float* __restrict__ inv,
                          const int* __restrict__ e0, const int* __restrict__ e1) {
  long long idx = (long long)blockIdx.x * blockDim.x + threadIdx.x;
  if (idx >= (long long)N_EDGES * 32) return;
  int e = (int)(idx >> 5);
  int c = ((int)idx & 31) * 4;
  int u = e0[e], v = e1[e];
  float cf = inv[u] * inv[v];
  float4 hu = *(const float4*)(h + (size_t)u * DIM + c);
  float4 hv = *(const float4*)(h + (size_t)v * DIM + c);
  float* av = agg + (size_t)v * DIM + c;
  float* au = agg + (size_t)u * DIM + c;
  atomicAdd(av + 0, hu.x * cf); atomicAdd(av + 1, hu.y * cf);
  atomicAdd(av + 2, hu.z * cf); atomicAdd(av + 3, hu.w * cf);
  atomicAdd(au + 0, hv.x * cf); atomicAdd(au + 1, hv.y * cf);
  atomicAdd(au + 2, hv.z * cf); atomicAdd(au + 3, hv.w * cf);
}

// hout = relu(agg @ W) + hin, one 16x16 tile per wave via V_WMMA_F32_16X16X4_F32.
// Block = 256 threads = 8 waves covering 16 rows x 128 cols.
__global__ __launch_bounds__(256) void k_gemm_relu_res(
    float* __restrict__ hout, const float* __restrict__ agg,
    const float* __restrict__ W, const float* __restrict__ hin) {
  int lane = threadIdx.x & 31;
  int wave = threadIdx.x >> 5;
  int row0 = blockIdx.x * 16;       // N divisible by 16
  int n0   = wave * 16;             // DIM/16 == 8 waves
  int half = lane >> 4;             // 0: K+0/K+1 slice, 1: K+2/K+3 slice
  int lm   = lane & 15;

  v8f acc = {};
  // A (16x4 f32 layout): lane<16 -> A[m, k+0],A[m, k+1]; lane>=16 -> A[m, k+2],A[m, k+3]
  const float* arow = agg + (size_t)(row0 + lm) * DIM + 2 * half;
  // B (4x16 f32 layout): lane<16 -> B[k+0, n],B[k+1, n]; lane>=16 -> B[k+2, n],B[k+3, n]
  const float* bcol = W + (size_t)(2 * half) * DIM + n0 + lm;

  for (int k = 0; k < DIM; k += 4) {
    v2f A, B;
    A.x = arow[k];
    A.y = arow[k + 1];
    B.x = bcol[(size_t)k * DIM];
    B.y = bcol[(size_t)(k + 1) * DIM];
    acc = __builtin_amdgcn_wmma_f32_16x16x4_f32(
        /*neg_a=*/false, A, /*neg_b=*/false, B,
        /*c_mod=*/(short)0, acc, /*reuse_a=*/false, /*reuse_b=*/false);
  }

  // C/D layout: VGPR r -> (M=r, N=lane) for lanes 0-15, (M=r+8, N=lane-16) for 16-31
  int col = n0 + lm;
  int mb  = half * 8;
  #pragma unroll
  for (int r = 0; r < 8; ++r) {
    size_t o = (size_t)(row0 + mb + r) * DIM + col;
    hout[o] = fmaxf(acc[r], 0.0f) + hin[o];
  }
}

__global__ void k_gather_seeds(float* __restrict__ out, const float* __restrict__ ent,
                               const int* __restrict__ seeds) {
  int idx = blockIdx.x * blockDim.x + threadIdx.x;  // S * (DIM/4) chunks
  if (idx < N_SEEDS * (DIM / 4)) {
    int s = idx >> 5;
    int c = (idx & 31) * 4;
    int node = seeds[s];
    *(float4*)(out + (size_t)s * DIM + c) =
        *(const float4*)(ent + (size_t)node * DIM + c);
  }
}

static void run_graph(const int* edges, const float* emb, const int* seeds,
                      const float* W0, const float* W1,
                      float* seed_out, float* ent_out,
                      float* deg, float* h1, float* agg, hipStream_t stream) {
  const int* e0 = edges;
  const int* e1 = edges + N_EDGES;

  k_deg_init<<<cdiv(N_NODES, 256), 256, 0, stream>>>(deg);
  k_deg_count<<<cdiv(N_EDGES, 256), 256, 0, stream>>>(e0, e1, deg);
  k_rsqrt<<<cdiv(N_NODES, 256), 256, 0, stream>>>(deg);  // deg now holds inv_sqrt

  // layer 0: h1 = relu(agg(emb) @ W0) + emb
  k_agg_init<<<cdiv((long long)N_NODES * 32, 256), 256, 0, stream>>>(agg, emb, deg);
  k_scatter<<<cdiv((long long)N_EDGES * 32, 256), 256, 0, stream>>>(agg, emb, deg, e0, e1);
  k_gemm_relu_res<<<N_NODES / 16, 256, 0, stream>>>(h1, agg, W0, emb);

  // layer 1: ent_out = relu(agg(h1) @ W1) + h1
  k_agg_init<<<cdiv((long long)N_NODES * 32, 256), 256, 0, stream>>>(agg, h1, deg);
  k_scatter<<<cdiv((long long)N_EDGES * 32, 256), 256, 0, stream>>>(agg, h1, deg, e0, e1);
  k_gemm_relu_res<<<N_NODES / 16, 256, 0, stream>>>(ent_out, agg, W1, h1);

  k_gather_seeds<<<cdiv((long long)N_SEEDS * 32, 256), 256, 0, stream>>>(seed_out, ent_out, seeds);
}

extern "C" void kernel_launch(void* const* d_in, const int* in_sizes, int n_in,
                              void* d_out, int out_size, void* d_ws, size_t ws_size,
                              hipStream_t stream) {
  const int*   sr_seeds = (const int*)d_in[0];
  const int*   tg_seeds = (const int*)d_in[1];
  const int*   edges_sr = (const int*)d_in[2];
  const int*   edges_tg = (const int*)d_in[3];
  const float* emb_sr   = (const float*)d_in[4];
  const float* emb_tg   = (const float*)d_in[5];
  const float* W0       = (const float*)d_in[6];
  const float* W1       = (const float*)d_in[7];

  float* out      = (float*)d_out;
  float* sr_seed  = out;
  float* tg_seed  = sr_seed + (size_t)N_SEEDS * DIM;
  float* sr_ent   = tg_seed + (size_t)N_SEEDS * DIM;
  float* tg_ent   = sr_ent + (size_t)N_NODES * DIM;

  float* deg = (float*)d_ws;                       // N floats (becomes inv_sqrt)
  float* h1  = deg + N_NODES;                      // N*D floats
  float* agg = h1 + (size_t)N_NODES * DIM;         // N*D floats

  run_graph(edges_sr, emb_sr, sr_seeds, W0, W1, sr_seed, sr_ent, deg, h1, agg, stream);
  run_graph(edges_tg, emb_tg, tg_seeds, W0, W1, tg_seed, tg_ent, deg, h1, agg, stream);
}